// CascadeCodebookCluster_90460601188596
// MI455X (gfx1250) — compile-verified
//
#include <hip/hip_runtime.h>
#include <hip/hip_bf16.h>
#include <float.h>

typedef float v2f __attribute__((ext_vector_type(2)));
typedef float v4f __attribute__((ext_vector_type(4)));
typedef float v8f __attribute__((ext_vector_type(8)));

#define D_DIM 256
#define A_STRIDE 132            // 128 data floats per (h,m) slot + 4 pad (bank spread)
#define WAVES_PER_BLOCK 4
#define THREADS (WAVES_PER_BLOCK * 32)
#define MT 2                    // M-tiles (16 rows each) per wave
#define ROWS_PER_WAVE (MT * 16)               // 32
#define ROWS_PER_BLOCK (WAVES_PER_BLOCK * ROWS_PER_WAVE)   // 128
#define A_SLOTS (WAVES_PER_BLOCK * MT * 32)   // 256 (h,m) slots

// ---------------------------------------------------------------------------
// Codebook row norms: norms[k] = sum_d cb[k][d]^2   (6174 rows total, trivial)
// ---------------------------------------------------------------------------
__global__ void cc_norms(const float* __restrict__ cb0, int K0,
                         const float* __restrict__ cb1, int K1,
                         const float* __restrict__ cb2, int K2,
                         float* __restrict__ norms) {
    int i = blockIdx.x * blockDim.x + threadIdx.x;
    int total = K0 + K1 + K2;
    if (i >= total) return;
    const float* src;
    int row;
    if (i < K0)           { src = cb0; row = i; }
    else if (i < K0 + K1) { src = cb1; row = i - K0; }
    else                  { src = cb2; row = i - K0 - K1; }
    const v4f* p = (const v4f*)(src + (size_t)row * D_DIM);
    float s = 0.f;
    for (int j = 0; j < D_DIM / 4; ++j) {
        v4f v = p[j];
        s += v[0]*v[0] + v[1]*v[1] + v[2]*v[2] + v[3]*v[3];
    }
    norms[i] = s;
}

// ---------------------------------------------------------------------------
// Main: per (128-row block, layer): WMMA distance tiles + argmin + gather + loss
// Each wave: two 16x16 accumulator tiles vs one shared 16-wide B tile.
// ---------------------------------------------------------------------------
__global__ __launch_bounds__(THREADS)
void cc_main(const float* __restrict__ embeds,
             const float* __restrict__ cb0, int K0,
             const float* __restrict__ cb1, int K1,
             const float* __restrict__ cb2, int K2,
             const float* __restrict__ norms,
             float* __restrict__ out,
             float* __restrict__ partials,
             int nRows) {
    extern __shared__ float smem[];
    float* ldsA = smem;                           // A_SLOTS * A_STRIDE floats
    float* ldsB = smem + A_SLOTS * A_STRIDE;      //      32 * A_STRIDE floats

    const int tid   = threadIdx.x;
    const int lane  = tid & 31;
    const int w     = tid >> 5;
    const int layer = blockIdx.y;

    const float* cb; int K; int normOff;
    if (layer == 0)      { cb = cb0; K = K0; normOff = 0; }
    else if (layer == 1) { cb = cb1; K = K1; normOff = K0; }
    else                 { cb = cb2; K = K2; normOff = K0 + K1; }

    const int rowBase = blockIdx.x * ROWS_PER_BLOCK;

    // ---- Stage A panel (128 rows x 256 f32) into WMMA A-operand layout ----
    // WMMA f32 16x16x4 A layout: lane L holds M=L%16, half h=L/16; VGPR r is
    // element K = 2h + r of the 4-wide K chunk. Each (h,m) slot is a
    // contiguous K-stream so compute reads are single ds_load_b128s.
    {
        int lr = tid;                             // one row per thread
        int wA = lr >> 5;                         // owning wave
        int st = (lr >> 4) & 1;                   // which of the wave's 2 M-tiles
        int mA = lr & 15;
        const v4f* src = (const v4f*)(embeds + (size_t)(rowBase + lr) * D_DIM);
        float* dst0 = ldsA + (wA * (MT * 32) + st * 32 + mA) * A_STRIDE;       // h=0
        float* dst1 = dst0 + 16 * A_STRIDE;                                    // h=1
#pragma unroll
        for (int i = 0; i < D_DIM / 4; ++i) {     // element d = 4i: (h0:j0,j1, h1:j0,j1)
            v4f v = src[i];
            *(v2f*)(dst0 + 2 * i) = (v2f){v[0], v[1]};
            *(v2f*)(dst1 + 2 * i) = (v2f){v[2], v[3]};
        }
    }

    float bestS[MT * 8];
    int   bestI[MT * 8];
#pragma unroll
    for (int r = 0; r < MT * 8; ++r) { bestS[r] = FLT_MAX; bestI[r] = 0x7fffffff; }

    const int nLane = lane & 15;
    const float* aP0 = ldsA + (w * (MT * 32) + lane) * A_STRIDE;
    const float* aP1 = aP0 + 32 * A_STRIDE;
    const float* bP  = ldsB + lane * A_STRIDE;

    const int numTiles = (K + 15) >> 4;
    for (int t = 0; t < numTiles; ++t) {
        const int tileBase = t << 4;
        __syncthreads();            // previous tile fully consumed (also covers A stage)
        // ---- Cooperative B-tile stage (16 codebook rows), same layout as A ----
        {
            int n = tid >> 3;
            int p = tid & 7;
            int srcRow = tileBase + n;
            if (srcRow >= K) srcRow = K - 1;      // clamp; masked via +inf norm
            const v4f* src = (const v4f*)(cb + (size_t)srcRow * D_DIM);
            float* dst0 = ldsB + n * A_STRIDE;
            float* dst1 = dst0 + 16 * A_STRIDE;
#pragma unroll
            for (int i = 0; i < 8; ++i) {
                int c = p * 8 + i;                // float4 index within the row
                v4f v = src[c];
                *(v2f*)(dst0 + 2 * c) = (v2f){v[0], v[1]};
                *(v2f*)(dst1 + 2 * c) = (v2f){v[2], v[3]};
            }
        }
        __syncthreads();

        // ---- 128 accumulating f32 WMMAs over D=256: two independent chains,
        //      B operand loaded once from LDS and reused by both A tiles ----
        v8f acc0 = {0.f,0.f,0.f,0.f,0.f,0.f,0.f,0.f};
        v8f acc1 = {0.f,0.f,0.f,0.f,0.f,0.f,0.f,0.f};
#pragma unroll
        for (int c2 = 0; c2 < 32; ++c2) {
            v4f bv  = *(const v4f*)(bP  + c2 * 4);
            v4f av0 = *(const v4f*)(aP0 + c2 * 4);
            v4f av1 = *(const v4f*)(aP1 + c2 * 4);
            v2f b0  = {bv[0],  bv[1]},  b1  = {bv[2],  bv[3]};
            v2f a00 = {av0[0], av0[1]}, a01 = {av0[2], av0[3]};
            v2f a10 = {av1[0], av1[1]}, a11 = {av1[2], av1[3]};
            acc0 = __builtin_amdgcn_wmma_f32_16x16x4_f32(false, a00, false, b0,
                                                         (short)0, acc0, false, false);
            acc1 = __builtin_amdgcn_wmma_f32_16x16x4_f32(false, a10, false, b0,
                                                         (short)0, acc1, false, false);
            acc0 = __builtin_amdgcn_wmma_f32_16x16x4_f32(false, a01, false, b1,
                                                         (short)0, acc0, false, false);
            acc1 = __builtin_amdgcn_wmma_f32_16x16x4_f32(false, a11, false, b1,
                                                         (short)0, acc1, false, false);
        }

        // score(m,n) = ||w_n||^2 - 2*dot  (||e||^2 constant per row -> dropped)
        const int gk = tileBase + nLane;
        const bool valid = gk < K;
        const float wn = valid ? norms[normOff + gk] : FLT_MAX;
#pragma unroll
        for (int r = 0; r < 8; ++r) {
            float s0 = wn - 2.0f * acc0[r];
            float s1 = wn - 2.0f * acc1[r];
            if (valid && s0 < bestS[r])     { bestS[r]     = s0; bestI[r]     = gk; }
            if (valid && s1 < bestS[8 + r]) { bestS[8 + r] = s1; bestI[8 + r] = gk; }
        }
    }

    // ---- argmin reduction across the 16-lane halves (C layout: VGPR r ->
    //      M=r (lanes 0-15) or M=r+8 (lanes 16-31); lane%16 = N column) ----
#pragma unroll
    for (int r = 0; r < MT * 8; ++r) {
        float s = bestS[r];
        int   i = bestI[r];
#pragma unroll
        for (int off = 8; off >= 1; off >>= 1) {
            float os = __shfl_xor(s, off, 32);
            int   oi = __shfl_xor(i, off, 32);
            if (os < s || (os == s && oi < i)) { s = os; i = oi; }  // first-index tie-break
        }
        bestS[r] = s; bestI[r] = i;
    }

    // ---- gather q = cb[idx], write output, accumulate (q-e)^2 ----
    const int h = lane >> 4;
    float localSum = 0.f;
#pragma unroll
    for (int st = 0; st < MT; ++st) {
#pragma unroll
        for (int r = 0; r < 8; ++r) {
            const int m   = r + 8 * h;
            const int row = rowBase + w * ROWS_PER_WAVE + st * 16 + m;
            const int idx = bestI[st * 8 + r];
            const v4f* qsrc = (const v4f*)(cb + (size_t)idx * D_DIM);
            const v4f* esrc = (const v4f*)(embeds + (size_t)row * D_DIM);
            v4f* dst = (v4f*)(out + ((size_t)layer * nRows + row) * D_DIM);
#pragma unroll
            for (int vv = 0; vv < 4; ++vv) {
                int q4 = nLane * 4 + vv;          // 16 floats per lane per row
                v4f q = qsrc[q4];
                v4f e = esrc[q4];
                dst[q4] = q;
                v4f df = q - e;
                localSum += df[0]*df[0] + df[1]*df[1] + df[2]*df[2] + df[3]*df[3];
            }
        }
    }

    // ---- deterministic per-block loss partial (serial thread-0 reduce) ----
    __syncthreads();
    ldsB[tid] = localSum;
    __syncthreads();
    if (tid == 0) {
        float s = 0.f;
        for (int i = 0; i < THREADS; ++i) s += ldsB[i];
        partials[blockIdx.y * gridDim.x + blockIdx.x] = s;
    }
}

// ---------------------------------------------------------------------------
// Fixed-order final sum -> q_loss = 1.25 * sum_layers mean((q-e)^2)
// ---------------------------------------------------------------------------
__global__ void cc_finalize(const float* __restrict__ partials, int n,
                            float* __restrict__ out, size_t lossIdx, float invND) {
    if (blockIdx.x == 0 && threadIdx.x == 0) {
        float s = 0.f;
        for (int i = 0; i < n; ++i) s += partials[i];
        out[lossIdx] = 1.25f * invND * s;
    }
}

extern "C" void kernel_launch(void* const* d_in, const int* in_sizes, int n_in,
                              void* d_out, int out_size, void* d_ws, size_t ws_size,
                              hipStream_t stream) {
    const float* embeds = (const float*)d_in[0];
    const float* cb0    = (const float*)d_in[1];
    const float* cb1    = (const float*)d_in[2];
    const float* cb2    = (const float*)d_in[3];

    const int nRows = in_sizes[0] / D_DIM;     // 16384
    const int K0 = in_sizes[1] / D_DIM;        // 5832
    const int K1 = in_sizes[2] / D_DIM;        // 324
    const int K2 = in_sizes[3] / D_DIM;        // 18
    const int totalK = K0 + K1 + K2;

    float* norms    = (float*)d_ws;            // totalK floats
    float* partials = (float*)d_ws + 8192;     // gx*3 floats

    cc_norms<<<(totalK + 255) / 256, 256, 0, stream>>>(cb0, K0, cb1, K1, cb2, K2, norms);

    const int gx = nRows / ROWS_PER_BLOCK;     // 128
    const size_t smemBytes =
        (size_t)(A_SLOTS + 32) * A_STRIDE * sizeof(float);   // ~148.7 KB
    cc_main<<<dim3(gx, 3), THREADS, smemBytes, stream>>>(
        embeds, cb0, K0, cb1, K1, cb2, K2, norms,
        (float*)d_out, partials, nRows);

    cc_finalize<<<1, 1, 0, stream>>>(partials, gx * 3, (float*)d_out,
                                     (size_t)out_size - 1,
                                     1.0f / ((float)nRows * (float)D_DIM));
}